// Head_15272903705216
// MI455X (gfx1250) — compile-verified
//
#include <hip/hip_runtime.h>

// Shapes (all static): x(1,6,16,16) f32, Wq(32,16), Wk(16,32,16), bk(16,32),
// Wv(16,32,16), bv(16,32)  ->  out(1,6,16,32) f32.
//
// Factorized algorithm (no KP/VP materialization):
//   q[m,e]    = pt_flat[m,:] . Wq[e,:]            (96x16 @ 16x32)
//   qk[i,j,d] = sum_e q[i,j,e] Wk[j][e,d]         (per-j 6x32 @ 32x16)
//   qb[i,j]   = q[i,j,:] . bk[j,:]
//   s[i,j,fn] = qk[i,j,:] . P_i[fn,:] + qb[i,j];  cs = CNT * s
//   u[i,j,:]  = sum_fn cs[i,j,fn] * P_i[fn,:];    t[i,j] = sum_fn cs
//   out[i,j,e]= sum_d Wv[j][e,d] u[i,j,d] + bv[j,e] * t[i,j]
// All GEMMs run on V_WMMA_F32_16X16X4_F32 (fp32-exact), one workgroup.

typedef float v2f __attribute__((ext_vector_type(2)));
typedef float v8f __attribute__((ext_vector_type(8)));

__device__ __forceinline__ v8f wmma_k4(v2f a, v2f b, v8f c) {
  // 8 args: (neg_a, A, neg_b, B, c_mod, C, reuse_a, reuse_b)
  return __builtin_amdgcn_wmma_f32_16x16x4_f32(false, a, false, b, (short)0, c,
                                               false, false);
}

// P_i[fn, d]: scrambled zero-padded context view, indexed straight out of pt.
// pt LDS layout: pt[(n*6 + c)*16 + d]  (n = patch 0..15, c = frame 0..5)
__device__ __forceinline__ float loadP(const float* s_pt, int i, int fn, int d) {
  const int ip1 = i + 1;
  if (fn >= ip1 * 16) return 0.0f;           // zero-padded frames (CNT==0 there)
  const int n = fn / ip1;
  const int c = fn - n * ip1;
  return s_pt[(n * 6 + c) * 16 + d];
}

__global__ __launch_bounds__(256) void lcattn_kernel(
    const float* __restrict__ x,  const float* __restrict__ Wq,
    const float* __restrict__ Wk, const float* __restrict__ bk,
    const float* __restrict__ Wv, const float* __restrict__ bv,
    float* __restrict__ out) {
  __shared__ float s_pt[1536];    // (16,6,16) patchified x; flat == q-scramble
  __shared__ float s_q[3072];     // (6,16,32)
  __shared__ float s_qk[1536];    // (6,16,16)
  __shared__ float s_qb[96];      // (6,16)
  __shared__ float s_cs[1536];    // (16,96) per-i scores * CNT
  __shared__ float s_u[1536];     // (6,16,16)
  __shared__ float s_t[96];       // (6,16)
  __shared__ unsigned char s_cnt[9216];  // CNT (6,16,6,16), small ints

  const int tid  = threadIdx.x;
  const int wave = tid >> 5;      // 8 waves of 32 (wave32)
  const int lane = tid & 31;
  const int col  = lane & 15;     // N-column / M-row within a 16x16 tile
  const int half = lane >> 4;     // lane half selects K pair (A) / M+8 (C,D)

  if (tid == 0) {                 // CDNA5 global_prefetch_b8
    __builtin_prefetch(x, 0, 0);
    __builtin_prefetch(Wk, 0, 0);
    __builtin_prefetch(Wv, 0, 0);
  }

  // ---- phase 0: patchify + build CNT on device -------------------------
  for (int idx = tid; idx < 1536; idx += 256) {
    const int d = idx & 15;
    const int rest = idx >> 4;
    const int c = rest % 6;
    const int n = rest / 6;
    const int hp = n >> 2, wp = n & 3;
    const int pr = d >> 2, pc = d & 3;
    s_pt[idx] = x[c * 256 + (hp * 4 + pr) * 16 + (wp * 4 + pc)];
  }
  if (tid < 96) {
    const int iq = tid >> 4, j = tid & 15;
    unsigned char* my = &s_cnt[(iq * 16 + j) * 96];
    for (int z = 0; z < 96; ++z) my[z] = 0;
    my[iq * 16 + j] += 1;                       // initial entry (f=iq, n=j)
    int il = 2;
    for (int fp = iq; fp >= 0; --fp) {
      for (int k = -il + 1; k <= il - 1; ++k)
        for (int l = -il + 1; l <= il - 1; ++l)
          if (!(j == 0 && l == 0 && il == 2)) {
            const int idx = j + 16 * k + l;
            if (idx >= 0 && idx < 16) my[fp * 16 + idx] += 1;
          }
      ++il;
    }
  }
  __syncthreads();

  // ---- phase 1: q (96x16 @ 16x32) -> 12 WMMA tiles ---------------------
  for (int tl = wave; tl < 12; tl += 8) {
    const int mt = tl >> 1, ct = tl & 1;
    v8f acc = {0.f, 0.f, 0.f, 0.f, 0.f, 0.f, 0.f, 0.f};
#pragma unroll
    for (int kk = 0; kk < 4; ++kk) {
      const int k0 = kk * 4 + half * 2;
      const int m = mt * 16 + col;
      v2f a = {s_pt[m * 16 + k0], s_pt[m * 16 + k0 + 1]};
      const int e = ct * 16 + col;
      v2f b = {Wq[e * 16 + k0], Wq[e * 16 + k0 + 1]};   // B[k][n] = Wq[n][k]
      acc = wmma_k4(a, b, acc);
    }
#pragma unroll
    for (int r = 0; r < 8; ++r) {
      const int m = mt * 16 + r + 8 * half;
      s_q[m * 32 + ct * 16 + col] = acc[r];
    }
  }
  __syncthreads();

  // ---- phase 2: qk (per-j, rows i<6 padded to 16), then qb -------------
  for (int j = wave; j < 16; j += 8) {
    v8f acc = {0.f, 0.f, 0.f, 0.f, 0.f, 0.f, 0.f, 0.f};
#pragma unroll
    for (int kk = 0; kk < 8; ++kk) {
      const int k0 = kk * 4 + half * 2;
      v2f a;
      a.x = (col < 6) ? s_q[(col * 16 + j) * 32 + k0] : 0.0f;
      a.y = (col < 6) ? s_q[(col * 16 + j) * 32 + k0 + 1] : 0.0f;
      v2f b = {Wk[j * 512 + k0 * 16 + col],              // B[e][d]=Wk[j][e][d]
               Wk[j * 512 + (k0 + 1) * 16 + col]};
      acc = wmma_k4(a, b, acc);
    }
#pragma unroll
    for (int r = 0; r < 8; ++r) {
      const int i = r + 8 * half;
      if (i < 6) s_qk[(i * 16 + j) * 16 + col] = acc[r];
    }
  }
  __syncthreads();
  if (tid < 96) {
    const int i = tid >> 4, j = tid & 15;
    float a = 0.f;
    for (int e = 0; e < 32; ++e) a += s_q[(i * 16 + j) * 32 + e] * bk[j * 32 + e];
    s_qb[tid] = a;
  }
  __syncthreads();

  // ---- phase 3: per-i scores + reduction -------------------------------
  for (int i = 0; i < 6; ++i) {
    if (wave < 6) {  // 3a: s_i = qk_i (16x16) @ P_i^T (16x96), tile ft=wave
      const int ft = wave;
      v8f acc = {0.f, 0.f, 0.f, 0.f, 0.f, 0.f, 0.f, 0.f};
#pragma unroll
      for (int kk = 0; kk < 4; ++kk) {
        const int k0 = kk * 4 + half * 2;
        v2f a = {s_qk[(i * 16 + col) * 16 + k0],
                 s_qk[(i * 16 + col) * 16 + k0 + 1]};
        v2f b = {loadP(s_pt, i, ft * 16 + col, k0),
                 loadP(s_pt, i, ft * 16 + col, k0 + 1)};
        acc = wmma_k4(a, b, acc);
      }
#pragma unroll
      for (int r = 0; r < 8; ++r) {
        const int j = r + 8 * half;
        const float sv = acc[r] + s_qb[i * 16 + j];
        s_cs[j * 96 + ft * 16 + col] =
            sv * (float)s_cnt[((i * 16 + j) * 6 + ft) * 16 + col];
      }
    }
    __syncthreads();
    if (wave == 0) {  // 3b: u_i (16x16) = cs (16x96) @ P_i (96x16), K=96
      v8f acc = {0.f, 0.f, 0.f, 0.f, 0.f, 0.f, 0.f, 0.f};
#pragma unroll
      for (int kk = 0; kk < 24; ++kk) {
        const int k0 = kk * 4 + half * 2;
        v2f a = {s_cs[col * 96 + k0], s_cs[col * 96 + k0 + 1]};
        v2f b = {loadP(s_pt, i, k0, col), loadP(s_pt, i, k0 + 1, col)};
        acc = wmma_k4(a, b, acc);
      }
#pragma unroll
      for (int r = 0; r < 8; ++r) {
        const int j = r + 8 * half;
        s_u[(i * 16 + j) * 16 + col] = acc[r];
      }
    } else if (wave == 1 && lane < 16) {  // t[i][j] = row-sum of cs
      float a = 0.f;
      for (int fn = 0; fn < 96; ++fn) a += s_cs[lane * 96 + fn];
      s_t[i * 16 + lane] = a;
    }
    __syncthreads();
  }

  // ---- phase 4: out (per-j 6x16 @ 16x32; bias*t folded into C init) ----
  for (int tl = wave; tl < 32; tl += 8) {
    const int j = tl >> 1, ct = tl & 1;
    const int e = ct * 16 + col;
    v8f acc;
#pragma unroll
    for (int r = 0; r < 8; ++r) {
      const int i = r + 8 * half;
      acc[r] = (i < 6) ? bv[j * 32 + e] * s_t[i * 16 + j] : 0.0f;
    }
#pragma unroll
    for (int kk = 0; kk < 4; ++kk) {
      const int k0 = kk * 4 + half * 2;
      v2f a;
      a.x = (col < 6) ? s_u[(col * 16 + j) * 16 + k0] : 0.0f;
      a.y = (col < 6) ? s_u[(col * 16 + j) * 16 + k0 + 1] : 0.0f;
      v2f b = {Wv[j * 512 + e * 16 + k0],                // B[d][e]=Wv[j][e][d]
               Wv[j * 512 + e * 16 + k0 + 1]};
      acc = wmma_k4(a, b, acc);
    }
#pragma unroll
    for (int r = 0; r < 8; ++r) {
      const int i = r + 8 * half;
      if (i < 6) out[(i * 16 + j) * 32 + e] = acc[r];
    }
  }
}

extern "C" void kernel_launch(void* const* d_in, const int* in_sizes, int n_in,
                              void* d_out, int out_size, void* d_ws,
                              size_t ws_size, hipStream_t stream) {
  (void)in_sizes; (void)n_in; (void)out_size; (void)d_ws; (void)ws_size;
  const float* x  = (const float*)d_in[0];
  const float* Wq = (const float*)d_in[1];
  const float* Wk = (const float*)d_in[2];
  const float* bk = (const float*)d_in[3];
  const float* Wv = (const float*)d_in[4];
  const float* bv = (const float*)d_in[5];
  float* o = (float*)d_out;
  hipLaunchKernelGGL(lcattn_kernel, dim3(1), dim3(256), 0, stream,
                     x, Wq, Wk, bk, Wv, bv, o);
}